// TripletLoss_4750233829521
// MI455X (gfx1250) — compile-verified
//
#include <hip/hip_runtime.h>
#include <math.h>

// ---------------------------------------------------------------------------
// Batch-hard triplet loss for MI455X (gfx1250, wave32).
// Gram matrix via V_WMMA_F32_16X16X4_F32 (exact fp32), fused mining on
// SQUARED distances (sqrt is monotone -> one sqrt per row at the end).
// 4-way j-tile register blocking = 4 independent WMMA accumulator chains.
// All B loads are immediate offsets off ONE base VGPR pair (j-tile stride
// 16 KB and K stride fit the 24-bit IOFFSET) -> no inner-loop address math,
// loads clause up and hide L2 latency. Deterministic reductions throughout.
// ---------------------------------------------------------------------------

typedef float v2f __attribute__((ext_vector_type(2)));
typedef float v8f __attribute__((ext_vector_type(8)));

#define N_SAMP   8192
#define DIM      256
#define MARGIN_F 0.6f
#define WEIGHT_F 1.0f
#define TILE     16
#define JBLK     4            // j-tiles per wave pass (independent WMMA chains)
#define WAVES    8
#define BLOCK    (WAVES * 32)
#define PITCH    (DIM + 2)    // pad 2 floats: row stride 258 -> 16 rows hit 32
                              // distinct LDS banks for 8B reads, stays 8B aligned

// ---------------- pass 1: per-row sum of squares ----------------------------
__global__ __launch_bounds__(256) void rows_sumsq(const float* __restrict__ x,
                                                  float* __restrict__ sq) {
    int row = blockIdx.x * blockDim.x + threadIdx.x;
    if (row >= N_SAMP) return;
    const float4* p = (const float4*)(x + (size_t)row * DIM);
    float s = 0.0f;
#pragma unroll 8
    for (int c = 0; c < DIM / 4; ++c) {
        float4 v = p[c];
        s += v.x * v.x + v.y * v.y + v.z * v.z + v.w * v.w;
    }
    sq[row] = s;
}

// ---------------- pass 2: fused Gram (WMMA f32) + batch-hard mining ---------
__global__ __launch_bounds__(BLOCK) void triplet_mine(
    const float* __restrict__ x, const int* __restrict__ lab,
    const float* __restrict__ sq, float* __restrict__ rowloss) {

    __shared__ float As[TILE][PITCH];     // A panel: 16 rows x 256 (padded)
    __shared__ float sqiS[TILE];
    __shared__ int   labiS[TILE];
    __shared__ float wap[WAVES][TILE];    // per-wave partial max(d2) positives
    __shared__ float wan[WAVES][TILE];    // per-wave partial min(d2) negatives

    const int i0   = blockIdx.x * TILE;
    const int tid  = threadIdx.x;
    const int lane = tid & 31;
    const int wave = tid >> 5;
    const int hi   = lane >> 4;           // lane-half (selects K pair)
    const int colL = lane & 15;           // this lane's column within a j-tile

    // Stage A panel (16 x 256 fp32 = 16 KB) into LDS: 256 threads x 4 float4.
    {
        const float4* src = (const float4*)(x + (size_t)i0 * DIM);
#pragma unroll
        for (int it = 0; it < 4; ++it) {
            int idx = tid + it * BLOCK;          // 0..1023 float4 slots
            int r   = idx >> 6;                  // 64 float4 per row
            int c   = idx & 63;
            float4 v = src[idx];
            *(float4*)&As[r][c * 4] = v;
        }
        if (tid < TILE) {
            sqiS[tid]  = sq[i0 + tid];
            labiS[tid] = lab[i0 + tid];
        }
    }
    __syncthreads();

    // Per-lane row constants for the 8 C rows this lane owns (M = r + 8*hi).
    float sqi[8]; int labi[8];
#pragma unroll
    for (int r = 0; r < 8; ++r) {
        int M = r + 8 * hi;
        sqi[r]  = sqiS[M];
        labi[r] = labiS[M];
    }
    // Running reductions on SQUARED distances.
    float maxp[8], minn[8];
#pragma unroll
    for (int r = 0; r < 8; ++r) { maxp[r] = 0.0f; minn[r] = 3.0e38f; }

    // A operand per lane: row (lane&15) of panel, K = k + v + 2*hi
    // (V_WMMA_F32_16X16X4_F32 A layout). B = X^T slice, same 8B pattern.
    const float* arow = &As[colL][2 * hi];

    for (int jg = wave; jg < N_SAMP / (TILE * JBLK); jg += WAVES) {
        const int j0 = jg * TILE * JBLK;

        // ONE base address for all JBLK tiles; tile/K selection via IOFFSET.
        const float* bbase = x + (size_t)(j0 + colL) * DIM + 2 * hi;

        float sqj[JBLK]; int labj[JBLK];
#pragma unroll
        for (int t = 0; t < JBLK; ++t) {
            sqj[t]  = sq[j0 + t * TILE + colL];
            labj[t] = lab[j0 + t * TILE + colL];
        }

        v8f acc[JBLK];
#pragma unroll
        for (int t = 0; t < JBLK; ++t) acc[t] = (v8f){};

#pragma unroll 4
        for (int k = 0; k < DIM; k += 4) {
            v2f a = *(const v2f*)(arow + k);       // ds_load, shared by 4 WMMAs
            v2f b[JBLK];
#pragma unroll
            for (int t = 0; t < JBLK; ++t)         // immediate-offset loads,
                b[t] = *(const v2f*)(bbase + t * (TILE * DIM) + k); // clause up
#pragma unroll
            for (int t = 0; t < JBLK; ++t)
                acc[t] = __builtin_amdgcn_wmma_f32_16x16x4_f32(
                    /*neg_a=*/false, a, /*neg_b=*/false, b[t],
                    /*c_mod=*/(short)0, acc[t],
                    /*reuse_a=*/false, /*reuse_b=*/false);
        }

        // Fused squared-distance + mining (no sqrt: monotone, defer to end).
#pragma unroll
        for (int t = 0; t < JBLK; ++t) {
#pragma unroll
            for (int r = 0; r < 8; ++r) {
                float d2 = fmaxf(sqi[r] + sqj[t] - 2.0f * acc[t][r], 1e-12f);
                bool pos = (labi[r] == labj[t]);
                maxp[r] = fmaxf(maxp[r], pos ? d2 : 0.0f);
                minn[r] = fminf(minn[r], pos ? 3.0e38f : d2);
            }
        }
    }

    // Reduce across the 16 lanes of each half (columns of the tile).
#pragma unroll
    for (int r = 0; r < 8; ++r) {
#pragma unroll
        for (int off = 1; off < 16; off <<= 1) {
            maxp[r] = fmaxf(maxp[r], __shfl_xor(maxp[r], off, 32));
            minn[r] = fminf(minn[r], __shfl_xor(minn[r], off, 32));
        }
    }
    if (colL == 0) {
#pragma unroll
        for (int r = 0; r < 8; ++r) {
            int M = r + 8 * hi;
            wap[wave][M] = maxp[r];
            wan[wave][M] = minn[r];
        }
    }
    __syncthreads();

    // Cross-wave reduce + single sqrt per row + hinge.
    if (tid < TILE) {
        float ap2 = wap[0][tid], an2 = wan[0][tid];
#pragma unroll
        for (int w = 1; w < WAVES; ++w) {
            ap2 = fmaxf(ap2, wap[w][tid]);
            an2 = fminf(an2, wan[w][tid]);
        }
        float ap = sqrtf(fmaxf(ap2, 1e-12f));   // diagonal guarantees a positive
        float an = sqrtf(fmaxf(an2, 1e-12f));
        rowloss[i0 + tid] = fmaxf(ap - an + MARGIN_F, 0.0f);
    }
}

// ---------------- pass 3: deterministic mean --------------------------------
__global__ __launch_bounds__(256) void final_reduce(
    const float* __restrict__ rowloss, float* __restrict__ out) {
    __shared__ float s[256];
    float v = 0.0f;
    for (int i = threadIdx.x; i < N_SAMP; i += 256) v += rowloss[i];
    s[threadIdx.x] = v;
    __syncthreads();
    for (int off = 128; off > 0; off >>= 1) {
        if (threadIdx.x < off) s[threadIdx.x] += s[threadIdx.x + off];
        __syncthreads();
    }
    if (threadIdx.x == 0) out[0] = s[0] * (WEIGHT_F / (float)N_SAMP);
}

// ---------------------------------------------------------------------------
extern "C" void kernel_launch(void* const* d_in, const int* in_sizes, int n_in,
                              void* d_out, int out_size, void* d_ws, size_t ws_size,
                              hipStream_t stream) {
    (void)in_sizes; (void)n_in; (void)out_size; (void)ws_size;
    const float* x   = (const float*)d_in[0];
    const int*   lab = (const int*)d_in[1];   // labels (integer -> const int*)

    float* sq      = (float*)d_ws;            // [N_SAMP]
    float* rowloss = sq + N_SAMP;             // [N_SAMP]

    rows_sumsq  <<<N_SAMP / 256, 256,   0, stream>>>(x, sq);
    triplet_mine<<<N_SAMP / TILE, BLOCK, 0, stream>>>(x, lab, sq, rowloss);
    final_reduce<<<1,            256,   0, stream>>>(rowloss, (float*)d_out);
}